// WTAGNNLayer_46213848105760
// MI455X (gfx1250) — compile-verified
//
#include <hip/hip_runtime.h>

// ---------------------------------------------------------------------------
// WTAGNN layer for gfx1250 (MI455X), f32 via V_WMMA_F32_16X16X4_F32.
// D = 64 everywhere.
// Weight tiles live in LDS in K-pair-interleaved float2 layout so every WMMA
// B-fragment is a single aligned ds_load_b64 into an even VGPR pair.
// ---------------------------------------------------------------------------

typedef __attribute__((ext_vector_type(2))) float v2f;
typedef __attribute__((ext_vector_type(8))) float v8f;

#define Dm 64
#define PSTR 80  // float2 stride per K-pair row (PSTR%32==16 -> conflict-free)
#define WFL (32 * PSTR * 2)  // floats per weight tile in LDS (20 KB)
#define SS 68    // LDS stride for 16x64 stage tiles (bank-conflict padding)

__device__ __forceinline__ v8f wmma4(v2f a, v2f b, v8f c) {
  // D = A(16x4,f32) * B(4x16,f32) + C(16x16,f32)
  return __builtin_amdgcn_wmma_f32_16x16x4_f32(
      /*neg_a=*/false, a, /*neg_b=*/false, b,
      /*c_mod=*/(short)0, c, /*reuse_a=*/false, /*reuse_b=*/false);
}

// Load 16 A-fragments (one 16x64 row-tile, this lane's row) from a row ptr.
// Layout (ISA 7.12.2, 32-bit A 16x4): lane&15 = M, lane>=16 -> K base 2.
// Components are contiguous -> single 8-byte loads.
__device__ __forceinline__ void load_a_frags(const float* Arow, int kb, v2f a[16]) {
#pragma unroll
  for (int s = 0; s < 16; ++s) {
    a[s].x = Arow[4 * s + kb];
    a[s].y = Arow[4 * s + kb + 1];
  }
}

// 16x64 += 16x64 * 64x64 ; B resident in LDS, K-pair interleaved:
//   Wlds[(kp*PSTR + col)*2 + {0,1}] = W[2*kp + {0,1}][col]
// Fragment for k-step s, lane half h: kp = 2s + h  -> one v2f (b64) load.
__device__ __forceinline__ void gemm16x64(const v2f a[16], const float* Wlds,
                                          int l16, int half, v8f acc[4]) {
#pragma unroll
  for (int ct = 0; ct < 4; ++ct) {
    v8f c = acc[ct];
#pragma unroll
    for (int s = 0; s < 16; ++s) {
      const int kp = 2 * s + half;
      v2f b = *(const v2f*)&Wlds[(kp * PSTR + ct * 16 + l16) * 2];
      c = wmma4(a[s], b, c);
    }
    acc[ct] = c;
  }
}

__device__ __forceinline__ void load_weight_lds(float* dst, const float* src,
                                                int tid, int nthreads) {
  for (int i = tid; i < Dm * Dm; i += nthreads) {
    int r = i >> 6, c = i & 63;
    dst[((r >> 1) * PSTR + c) * 2 + (r & 1)] = src[i];
  }
}

// ---------------------------------------------------------------------------
// Pass A: nf_out = relu(nf@W_node + bias_node)  -> d_out
//         nfW    = nf_out @ Wbot                -> ws
// 128 threads = 4 waves, each wave owns a 16-row tile.
// ---------------------------------------------------------------------------
__global__ void __launch_bounds__(128) node_kernel(
    const float* __restrict__ nf, const float* __restrict__ W_node,
    const float* __restrict__ bias_node, const float* __restrict__ Wbot,
    float* __restrict__ nf_out, float* __restrict__ nfW, int N) {
  __shared__ float sWn[WFL];
  __shared__ float sWb[WFL];
  __shared__ float stage[4][16 * SS];

  const int tid = threadIdx.x;
  load_weight_lds(sWn, W_node, tid, 128);
  load_weight_lds(sWb, Wbot, tid, 128);
  __syncthreads();

  const int wave = tid >> 5, lane = tid & 31;
  const int half = lane >> 4, l16 = lane & 15, kb = half * 2;
  const int row0 = (blockIdx.x * 4 + wave) * 16;

  int m = row0 + l16;
  int mc = m < N ? m : N - 1;

  v2f a[16];
  load_a_frags(nf + (size_t)mc * Dm, kb, a);
  v8f acc[4] = {};
  gemm16x64(a, sWn, l16, half, acc);

  float* st = stage[wave];
#pragma unroll
  for (int ct = 0; ct < 4; ++ct) {
    int col = ct * 16 + l16;
    float b = bias_node[col];
#pragma unroll
    for (int r = 0; r < 8; ++r) {
      int M = r + 8 * half;
      float v = acc[ct][r] + b;
      v = v > 0.f ? v : 0.f;
      st[M * SS + col] = v;
      int row = row0 + M;
      if (row < N) nf_out[(size_t)row * Dm + col] = v;
    }
  }

  // chained GEMM: relu'd tile @ Wbot  (per-wave LDS region, no barrier needed)
  v2f a2[16];
  load_a_frags(st + l16 * SS, kb, a2);
  v8f acc2[4] = {};
  gemm16x64(a2, sWb, l16, half, acc2);
#pragma unroll
  for (int ct = 0; ct < 4; ++ct) {
    int col = ct * 16 + l16;
#pragma unroll
    for (int r = 0; r < 8; ++r) {
      int row = row0 + r + 8 * half;
      if (row < N) nfW[(size_t)row * Dm + col] = acc2[ct][r];
    }
  }
}

// ---------------------------------------------------------------------------
// Pass B: per 16-edge tile:
//   ef_h = ef@W_edge (registers); atomic nb_sum[dst]+=ef_h, deg[dst]+=1
//   efW2 = ef_h@Wtop -> written into d_out's ef region (in-place scratch)
// ---------------------------------------------------------------------------
__global__ void __launch_bounds__(128) edge_kernel(
    const float* __restrict__ ef, const int* __restrict__ dst,
    const float* __restrict__ W_edge, const float* __restrict__ Wtop,
    float* __restrict__ efW2, float* __restrict__ nb_sum,
    float* __restrict__ deg, int E) {
  __shared__ float sWe[WFL];
  __shared__ float sWt[WFL];
  __shared__ float stage[4][16 * SS];

  const int tid = threadIdx.x;
  load_weight_lds(sWe, W_edge, tid, 128);
  load_weight_lds(sWt, Wtop, tid, 128);
  __syncthreads();

  const int wave = tid >> 5, lane = tid & 31;
  const int half = lane >> 4, l16 = lane & 15, kb = half * 2;
  const long e0 = (long)(blockIdx.x * 4 + wave) * 16;

  long er = e0 + l16;
  if (er >= E) er = E - 1;

  v2f a[16];
  load_a_frags(ef + er * Dm, kb, a);
  v8f acc[4] = {};
  gemm16x64(a, sWe, l16, half, acc);

  // destination node ids for this lane's 8 output rows
  int dstr[8];
#pragma unroll
  for (int r = 0; r < 8; ++r) {
    long em = e0 + r + 8 * half;
    dstr[r] = (em < E) ? dst[em] : -1;
  }
  if (lane < 16) {
    long em = e0 + lane;
    if (em < E) atomicAdd(&deg[dst[em]], 1.0f);
  }

  float* st = stage[wave];
#pragma unroll
  for (int ct = 0; ct < 4; ++ct) {
    int col = ct * 16 + l16;
#pragma unroll
    for (int r = 0; r < 8; ++r) {
      int M = r + 8 * half;
      float v = acc[ct][r];
      st[M * SS + col] = v;
      if (dstr[r] >= 0) atomicAdd(&nb_sum[(size_t)dstr[r] * Dm + col], v);
    }
  }

  // chained GEMM: ef_h tile @ Wtop
  v2f a2[16];
  load_a_frags(st + l16 * SS, kb, a2);
  v8f acc2[4] = {};
  gemm16x64(a2, sWt, l16, half, acc2);
#pragma unroll
  for (int ct = 0; ct < 4; ++ct) {
    int col = ct * 16 + l16;
#pragma unroll
    for (int r = 0; r < 8; ++r) {
      long em = e0 + r + 8 * half;
      if (em < E) efW2[em * Dm + col] = acc2[ct][r];
    }
  }
}

// ---------------------------------------------------------------------------
// Pass C: nbW = (nb_sum / max(deg,1)) @ Wtop     (N-sized, 8 waves/block)
// ---------------------------------------------------------------------------
__global__ void __launch_bounds__(256) nbw_kernel(
    const float* __restrict__ nb_sum, const float* __restrict__ deg,
    const float* __restrict__ Wtop, float* __restrict__ nbW, int N) {
  __shared__ float sWt[WFL];
  const int tid = threadIdx.x;
  load_weight_lds(sWt, Wtop, tid, 256);
  __syncthreads();

  const int wave = tid >> 5, lane = tid & 31;
  const int half = lane >> 4, l16 = lane & 15, kb = half * 2;
  const int row0 = (blockIdx.x * 8 + wave) * 16;

  int m = row0 + l16;
  int mc = m < N ? m : N - 1;
  float dg = deg[mc];
  float inv = 1.0f / (dg > 1.0f ? dg : 1.0f);

  const float* Ar = nb_sum + (size_t)mc * Dm;
  v2f a[16];
#pragma unroll
  for (int s = 0; s < 16; ++s) {
    a[s].x = Ar[4 * s + kb] * inv;
    a[s].y = Ar[4 * s + kb + 1] * inv;
  }
  v8f acc[4] = {};
  gemm16x64(a, sWt, l16, half, acc);
#pragma unroll
  for (int ct = 0; ct < 4; ++ct) {
    int col = ct * 16 + l16;
#pragma unroll
    for (int r = 0; r < 8; ++r) {
      int row = row0 + r + 8 * half;
      if (row < N) nbW[(size_t)row * Dm + col] = acc[ct][r];
    }
  }
}

// ---------------------------------------------------------------------------
// Pass D: in-place finalize of ef_out:
//   ef_out = relu(efW2 + nbW[dst] + 0.5*(nfW[src]+nfW[dst]) + b_dense + bias_edge)
// Pure bandwidth; float4 per thread; gathers hit L2 (nbW/nfW ~25MB resident).
// ---------------------------------------------------------------------------
__global__ void __launch_bounds__(256) finalize_kernel(
    float* __restrict__ ef_out, const int* __restrict__ src,
    const int* __restrict__ dst, const float* __restrict__ nbW,
    const float* __restrict__ nfW, const float* __restrict__ b_dense,
    const float* __restrict__ bias_edge, long E) {
  long idx = (long)blockIdx.x * blockDim.x + threadIdx.x;  // float4 index
  long total = E * (Dm / 4);
  if (idx >= total) return;
  long e = idx >> 4;
  int q = ((int)idx & 15) * 4;
  int s = src[e], d = dst[e];

  const float4 v  = *(const float4*)(ef_out + e * Dm + q);
  const float4 nb = *(const float4*)(nbW + (size_t)d * Dm + q);
  const float4 fs = *(const float4*)(nfW + (size_t)s * Dm + q);
  const float4 fd = *(const float4*)(nfW + (size_t)d * Dm + q);
  const float4 bd = *(const float4*)(b_dense + q);
  const float4 be = *(const float4*)(bias_edge + q);

  float4 o;
  o.x = v.x + nb.x + 0.5f * (fs.x + fd.x) + bd.x + be.x;
  o.y = v.y + nb.y + 0.5f * (fs.y + fd.y) + bd.y + be.y;
  o.z = v.z + nb.z + 0.5f * (fs.z + fd.z) + bd.z + be.z;
  o.w = v.w + nb.w + 0.5f * (fs.w + fd.w) + bd.w + be.w;
  o.x = o.x > 0.f ? o.x : 0.f;
  o.y = o.y > 0.f ? o.y : 0.f;
  o.z = o.z > 0.f ? o.z : 0.f;
  o.w = o.w > 0.f ? o.w : 0.f;
  *(float4*)(ef_out + e * Dm + q) = o;
}

// ---------------------------------------------------------------------------
extern "C" void kernel_launch(void* const* d_in, const int* in_sizes, int n_in,
                              void* d_out, int out_size, void* d_ws,
                              size_t ws_size, hipStream_t stream) {
  const float* nf        = (const float*)d_in[0];
  const float* ef        = (const float*)d_in[1];
  const int*   src       = (const int*)d_in[2];
  const int*   dst       = (const int*)d_in[3];
  const float* W_node    = (const float*)d_in[4];
  const float* W_edge    = (const float*)d_in[5];
  const float* bias_node = (const float*)d_in[6];
  const float* bias_edge = (const float*)d_in[7];
  const float* W_dense   = (const float*)d_in[8];
  const float* b_dense   = (const float*)d_in[9];

  const int N = in_sizes[0] / Dm;
  const int E = in_sizes[1] / Dm;

  float* out    = (float*)d_out;
  float* nf_out = out;                     // [N,64]
  float* ef_out = out + (size_t)N * Dm;    // [E,64] (also efW2 scratch)

  float* ws     = (float*)d_ws;
  float* nb_sum = ws;                                  // [N,64]
  float* deg    = nb_sum + (size_t)N * Dm;             // [N]
  float* nfW    = deg + N;                             // [N,64]
  float* nbW    = nfW + (size_t)N * Dm;                // [N,64]

  const float* Wtop = W_dense;             // rows 0..63  of [128,64]
  const float* Wbot = W_dense + Dm * Dm;   // rows 64..127

  // zero the atomic accumulators (nb_sum + deg are contiguous)
  hipMemsetAsync(nb_sum, 0, sizeof(float) * ((size_t)N * Dm + N), stream);

  const int ntiles  = (N + 15) / 16;
  const int nblocks = (ntiles + 3) / 4;
  node_kernel<<<nblocks, 128, 0, stream>>>(nf, W_node, bias_node, Wbot, nf_out,
                                           nfW, N);

  const int etiles  = (E + 15) / 16;
  const int eblocks = (etiles + 3) / 4;
  edge_kernel<<<eblocks, 128, 0, stream>>>(ef, dst, W_edge, Wtop, ef_out,
                                           nb_sum, deg, E);

  const int nbblocks = (ntiles + 7) / 8;
  nbw_kernel<<<nbblocks, 256, 0, stream>>>(nb_sum, deg, Wtop, nbW, N);

  const long total   = (long)E * (Dm / 4);
  const int  fblocks = (int)((total + 255) / 256);
  finalize_kernel<<<fblocks, 256, 0, stream>>>(ef_out, src, dst, nbW, nfW,
                                               b_dense, bias_edge, (long)E);
}